// GNN_LGConv_homogen_variable_64244120814278
// MI455X (gfx1250) — compile-verified
//
#include <hip/hip_runtime.h>

// CDNA5 / gfx1250, wave32. LGConv + edge scoring.
// Phases: zero ws -> degree atomics -> rsqrt -> normalized scatter-add ->
//         WMMA (f32 16x16x4) batched edge dot-products.

typedef float v2f __attribute__((ext_vector_type(2)));
typedef float v8f __attribute__((ext_vector_type(8)));

__global__ void zero_ws_kernel(float* __restrict__ p, int n) {
    int i = blockIdx.x * blockDim.x + threadIdx.x;
    if (i < n) p[i] = 0.0f;
}

// deg[col[e]] += 1  (in-degree at targets), f32 counts are exact for deg < 2^24
__global__ void degree_kernel(const int* __restrict__ col, float* __restrict__ deg, int E) {
    int e = blockIdx.x * blockDim.x + threadIdx.x;
    if (e < E) unsafeAtomicAdd(&deg[col[e]], 1.0f);
}

// in-place: deg -> (deg > 0 ? rsqrt(deg) : 0)
__global__ void rsqrt_kernel(float* __restrict__ deg, int N) {
    int i = blockIdx.x * blockDim.x + threadIdx.x;
    if (i < N) {
        float d = deg[i];
        deg[i] = (d > 0.0f) ? __frsqrt_rn(d) : 0.0f;
    }
}

// One wave32 per edge. Edge metadata is wave-uniform -> scalar loads.
// Each lane handles 2 consecutive floats of the 64-wide row (coalesced 256B).
__global__ void scatter_kernel(const float* __restrict__ x,
                               const int* __restrict__ row,
                               const int* __restrict__ col,
                               const float* __restrict__ dis,
                               float* __restrict__ xout, int E) {
    int wave = (blockIdx.x * blockDim.x + threadIdx.x) >> 5;
    int lane = threadIdx.x & 31;
    if (wave >= E) return;
    int r = row[wave];
    int c = col[wave];
    float norm = dis[r] * dis[c];
    const float* xr = x + (size_t)r * 64 + lane * 2;
    float*       oc = xout + (size_t)c * 64 + lane * 2;
    v2f v = *(const v2f*)xr;
    unsafeAtomicAdd(&oc[0], norm * v.x);
    unsafeAtomicAdd(&oc[1], norm * v.y);
}

// 16 edge dot-products per wave via chained V_WMMA_F32_16X16X4_F32.
// A: rows x[ei0[e_base+m]]  (16x64)   B: cols x[ei1[e_base+n]]  (64x16)
// A layout (ISA 7.12.2, 32-bit A 16x4): lane m / m+16 holds K pair (0,1)/(2,3).
// B layout mirrors A with N = lane%16. Each lane therefore streams exactly half
// of one A-row and half of one B-row as float2 at 16B stride: no redundant bytes.
// Diagonal of D = the 16 dot products.
__global__ void wmma_dot_kernel(const float* __restrict__ x,
                                const int* __restrict__ ei,  // [2, Es] flat
                                float* __restrict__ logits, int Es) {
    int lane = threadIdx.x & 31;
    int wave = blockIdx.x * (blockDim.x >> 5) + (threadIdx.x >> 5);
    int e_base = wave * 16;
    if (e_base >= Es) return;                 // wave-uniform: EXEC stays all-1s

    int m  = lane & 15;
    int hi = (lane >> 4) & 1;                 // lane-half selects K offset +2
    int e  = e_base + m;
    if (e >= Es) e = Es - 1;                  // safe clamp (Es % 16 == 0 here)
    int ai = ei[e];                           // source row of edge e
    int bi = ei[Es + e];                      // dest   row of edge e

    const float* ap = x + (size_t)ai * 64 + 2 * hi;
    const float* bp = x + (size_t)bi * 64 + 2 * hi;

    v2f A[16], B[16];
#pragma unroll
    for (int k = 0; k < 16; ++k) {
        A[k] = *(const v2f*)(ap + 4 * k);
        B[k] = *(const v2f*)(bp + 4 * k);
    }

    v8f c0 = {}, c1 = {};                     // two accumulators for ILP
#pragma unroll
    for (int k = 0; k < 16; k += 2) {
        c0 = __builtin_amdgcn_wmma_f32_16x16x4_f32(false, A[k],     false, B[k],
                                                   (short)0, c0, false, false);
        c1 = __builtin_amdgcn_wmma_f32_16x16x4_f32(false, A[k + 1], false, B[k + 1],
                                                   (short)0, c1, false, false);
    }
    v8f c = c0 + c1;

    // Diagonal extraction per C/D layout: (m,m) m<8 -> lane m, vgpr m;
    // (8+j,8+j) -> lane 24+j, vgpr j.
    int idx = (lane < 8) ? lane : (lane - 24);
    float val = c[0];
#pragma unroll
    for (int j = 1; j < 8; ++j)
        if (idx == j) val = c[j];

    if (lane < 8) {
        int eo = e_base + lane;
        if (eo < Es) logits[eo] = val;
    } else if (lane >= 24) {
        int eo = e_base + (lane - 16);
        if (eo < Es) logits[eo] = val;
    }
}

extern "C" void kernel_launch(void* const* d_in, const int* in_sizes, int n_in,
                              void* d_out, int out_size, void* d_ws, size_t ws_size,
                              hipStream_t stream) {
    const float* x  = (const float*)d_in[0];   // [N, 64]
    // d_in[1] = W, d_in[2] = b : dead code in the reference, skipped.
    const int*   ei = (const int*)d_in[3];     // [2, Es] scoring edges
    const int*   pe = (const int*)d_in[4];     // [2, E]  propagation edges

    const int N  = in_sizes[0] / 64;
    const int Es = in_sizes[3] / 2;
    const int E  = in_sizes[4] / 2;

    float* deg  = (float*)d_ws;        // N floats; becomes deg^-1/2 in place
    float* xout = deg + N;             // N*64 floats

    const int zn = N * 65;
    zero_ws_kernel<<<(zn + 255) / 256, 256, 0, stream>>>(deg, zn);

    degree_kernel<<<(E + 255) / 256, 256, 0, stream>>>(pe + E, deg, E);

    rsqrt_kernel<<<(N + 255) / 256, 256, 0, stream>>>(deg, N);

    // one wave per edge -> E*32 threads
    {
        long long threads = (long long)E * 32;
        int blocks = (int)((threads + 255) / 256);
        scatter_kernel<<<blocks, 256, 0, stream>>>(x, pe, pe + E, deg, xout, E);
    }

    // 16 edges per wave, 8 waves per 256-thread block
    {
        int waves  = (Es + 15) / 16;
        int blocks = (waves + 7) / 8;
        wmma_dot_kernel<<<blocks, 256, 0, stream>>>(xout, ei, (float*)d_out, Es);
    }
}